// NeuralNetwork_42975442764251
// MI455X (gfx1250) — compile-verified
//
#include <hip/hip_runtime.h>

// ---------------------------------------------------------------------------
// Reference: A = I + a0*shift(844) + a1*shift(1264), L = 8137.
// U is nilpotent (U^10 == 0), so inv(A) = sum_k (-1)^k U^k exactly, and every
// reachable offset d = 844a + 1264b has a unique (a,b).  Hence
//   x[0,0,j] = sum_p c_p * q[0,0, j - d_p],   c_p = (-1)^(a+b) C(a+b,a) a0^a a1^b
// for the 40 feasible (a,b) pairs, and loss = sum_{j=L-1264}^{L-1} x[j]^2.
// 79 wave-sized 16-row tiles via V_WMMA_F32_16X16X4_F32 (4 taps / step,
// 12 steps over 48 padded taps).  Coefficients folded into the A operand;
// B is all-ones (immune to B-operand VGPR layout).  Fully branchless body.
// ---------------------------------------------------------------------------

typedef __attribute__((ext_vector_type(2))) float v2f;
typedef __attribute__((ext_vector_type(8))) float v8f;

#define LENGTH 8137
#define T1V    1264
#define JBASE  (LENGTH - T1V)   /* 6873 */
#define NTAP   48               /* 40 real taps + 8 zero pads */
#define NCHUNK (NTAP / 4)       /* 12 wmma steps */
#define NBLOCK (T1V / 16)       /* 79 waves, exact */

// Compile-time tap offsets d = 844a + 1264b (pads at the end are 0).
constexpr int TD[NTAP] = {
    0,844,1688,2532,3376,4220,5064,5908,6752,7596,
    1264,2108,2952,3796,4640,5484,6328,7172,8016,
    2528,3372,4216,5060,5904,6748,7592,
    3792,4636,5480,6324,7168,8012,
    5056,5900,6744,7588,
    6320,7164,8008,
    7584,
    0,0,0,0,0,0,0,0 };

// (a, b) exponents and signed binomial (-1)^(a+b)*C(a+b,a); used once at start.
__constant__ int TAP_A[NTAP] = {
    0,1,2,3,4,5,6,7,8,9,
    0,1,2,3,4,5,6,7,8,
    0,1,2,3,4,5,6,
    0,1,2,3,4,5,
    0,1,2,3,
    0,1,2,
    0,
    0,0,0,0,0,0,0,0 };

__constant__ int TAP_B[NTAP] = {
    0,0,0,0,0,0,0,0,0,0,
    1,1,1,1,1,1,1,1,1,
    2,2,2,2,2,2,2,
    3,3,3,3,3,3,
    4,4,4,4,
    5,5,5,
    6,
    0,0,0,0,0,0,0,0 };

__constant__ float TAP_S[NTAP] = {
     1.f,-1.f, 1.f,-1.f, 1.f,-1.f, 1.f,-1.f, 1.f,-1.f,
    -1.f, 2.f,-3.f, 4.f,-5.f, 6.f,-7.f, 8.f,-9.f,
     1.f,-3.f, 6.f,-10.f,15.f,-21.f,28.f,
    -1.f, 4.f,-10.f,20.f,-35.f,56.f,
     1.f,-5.f,15.f,-35.f,
    -1.f, 6.f,-21.f,
     1.f,
     0.f,0.f,0.f,0.f,0.f,0.f,0.f,0.f };

// Branchless c = s * a0^na * a1^nb (na <= 9, nb <= 6): fixed-trip select-muls.
__device__ __forceinline__ float tap_coeff(int p, float a0, float a1)
{
    const int na = TAP_A[p];
    const int nb = TAP_B[p];
    float c = TAP_S[p];
#pragma unroll
    for (int i = 0; i < 9; ++i) c = (i < na) ? c * a0 : c;
#pragma unroll
    for (int i = 0; i < 6; ++i) c = (i < nb) ? c * a1 : c;
    return c;
}

__global__ __launch_bounds__(32)
void fir_wmma_kernel(const float* __restrict__ q,
                     const float* __restrict__ pa0,
                     const float* __restrict__ pa1,
                     float* __restrict__ partial)
{
    __shared__ float lds_c[NTAP];
    const int lane = threadIdx.x;           // wave32: one wave per block
    const float a0 = pa0[0];
    const float a1 = pa1[0];

    // One-time: tap coefficients into LDS (48 = 32 + 16).
    lds_c[lane] = tap_coeff(lane, a0, a1);
    if (lane < NTAP - 32) lds_c[lane + 32] = tap_coeff(lane + 32, a0, a1);
    __syncthreads();

    // A-operand layout (documented): lane l holds row M = l&15;
    // VGPR v holds K = 2*(l>>4) + v.
    const int m    = lane & 15;
    const int half = lane >> 4;
    const int jm   = JBASE + (blockIdx.x << 4) + m;   // output index of row m

    // Phase 1: build all 12 A operands, fully branchless.
    v2f aops[NCHUNK];
#pragma unroll
    for (int t = 0; t < NCHUNK; ++t) {
#pragma unroll
        for (int v = 0; v < 2; ++v) {
            // Both candidate taps are compile-time: select by half (v_cndmask).
            const int d   = half ? TD[4 * t + 2 + v] : TD[4 * t + v];
            const int idx = jm - d;                        // >= -1143, <= 8136
            const float c  = lds_c[4 * t + 2 * half + v];  // unconditional ds_load
            const float qv = q[idx >= 0 ? idx : 0];        // clamped: always in-bounds
            aops[t][v] = (idx >= 0) ? qv * c : 0.0f;       // mul + v_cndmask
        }
    }

    // Phase 2: 12-deep WMMA accumulate chain.
    // D[m,n] += sum_k A[m,k] * 1  == sum_k c_k * q[j_m - d_k]  (all n equal)
    const v2f ones = {1.0f, 1.0f};   // B = all-ones: immune to B VGPR layout
    v8f acc = {};
#pragma unroll
    for (int t = 0; t < NCHUNK; ++t) {
        acc = __builtin_amdgcn_wmma_f32_16x16x4_f32(
                  false, aops[t], false, ones, (short)0, acc, false, false);
    }

    // D layout: lane 0 holds column N=0 for M=0..7 (VGPRs 0..7),
    // lane 16 holds M=8..15.  Emit per-half partial sum of squares.
    if (m == 0) {
        float s = 0.0f;
#pragma unroll
        for (int r = 0; r < 8; ++r) s += acc[r] * acc[r];
        partial[(blockIdx.x << 1) + half] = s;
    }
}

__global__ void loss_reduce_kernel(const float* __restrict__ partial,
                                   float* __restrict__ out, int n)
{
    if (threadIdx.x == 0 && blockIdx.x == 0) {
        float s = 0.0f;
        for (int i = 0; i < n; ++i) s += partial[i];   // fixed order: deterministic
        out[0] = s;
    }
}

extern "C" void kernel_launch(void* const* d_in, const int* in_sizes, int n_in,
                              void* d_out, int out_size, void* d_ws, size_t ws_size,
                              hipStream_t stream)
{
    (void)in_sizes; (void)n_in; (void)out_size; (void)ws_size;
    const float* q  = (const float*)d_in[0];   // [4, 8, 8137]; row (0,0) = q[0..8136]
    const float* a0 = (const float*)d_in[1];   // scalar
    const float* a1 = (const float*)d_in[2];   // scalar
    float* out      = (float*)d_out;           // 1 float
    float* part     = (float*)d_ws;            // 2 * NBLOCK floats

    fir_wmma_kernel<<<NBLOCK, 32, 0, stream>>>(q, a0, a1, part);
    loss_reduce_kernel<<<1, 32, 0, stream>>>(part, out, 2 * NBLOCK);
}